// SeparationLoss_46866683133955
// MI455X (gfx1250) — compile-verified
//
#include <hip/hip_runtime.h>
#include <hip/hip_bf16.h>
#include <math.h>

// Problem dims (from reference): inputs [32, 8, 512, 24, 12] fp32
#define NB   32
#define NP   8
#define NC   512
#define NHW  288          // 24*12
#define NBP  (NB * NP)    // 256
#define NTIL (NHW / 16)   // 18 column tiles of 16
#define PFD  64           // prefetch distance in channels
#define NCM  (NC - PFD)   // main-loop bound (448)

#define EPSF 1e-12f

typedef __attribute__((ext_vector_type(2))) float v2f;
typedef __attribute__((ext_vector_type(8))) float v8f;

// ---------------------------------------------------------------------------
// Kernel A: per (b,p) slab, compute f[hw] = sum_c x^2 via V_WMMA_F32_16X16X4_F32
// with an all-ones A matrix (D = 1 x B + C => every D row = column sum of B).
// Then L2-normalize f over hw in LDS and emit normalized f + its sum.
// Grid: 256 blocks (one per (b,p)), 128 threads (4 waves).
// ---------------------------------------------------------------------------
__global__ __launch_bounds__(128) void sep_sumsq_norm_kernel(
    const float* __restrict__ x,   // [NBP, NC, NHW]
    float* __restrict__ fn_out,    // [NBP, NHW] normalized f
    float* __restrict__ sfn_out)   // [NBP] sum of normalized f
{
    __shared__ float lds_f[NHW];
    __shared__ float red1[128];
    __shared__ float red2[128];
    __shared__ float inv_lds;

    const int tid    = threadIdx.x;
    const int waveId = tid >> 5;
    const int lane   = tid & 31;
    const int half   = lane >> 4;    // which 16-lane half of the wave
    const int nsub   = lane & 15;    // column within the 16-wide tile
    const int bp     = blockIdx.x;

    const float* base = x + (size_t)bp * (size_t)(NC * NHW);

    // A matrix = all ones (16x4 f32 -> 2 VGPRs per lane, every slot 1.0)
    v2f ones;
    ones[0] = 1.0f;
    ones[1] = 1.0f;

    // Each wave owns column-tiles waveId, waveId+4, ...
    for (int tile = waveId; tile < NTIL; tile += 4) {
        const int col = tile * 16 + nsub;
        // Per-lane streaming pointer: channel (c + half), column col.
        const float* p = base + (size_t)half * NHW + col;

        // Two independent accumulators to break the WMMA->WMMA RAW chain.
        v8f accA = {};
        v8f accB = {};

        // Main loop: unconditional prefetch PFD channels ahead (branch-free).
        #pragma unroll 4
        for (int c0 = 0; c0 < NCM; c0 += 8) {
            __builtin_prefetch(p + (size_t)(c0 + PFD) * NHW, 0, 3);
            // Chunk 1: channels c0..c0+3 for 16 columns.
            // Lane (half,nsub): B vgpr0 <- ch c0+half, B vgpr1 <- ch c0+2+half.
            // Each column's 4 K-slots get exactly channels {c0..c0+3} (any
            // permutation is fine: A is all-ones, we only need the K-sum).
            float a0 = p[(size_t)(c0    ) * NHW];
            float a1 = p[(size_t)(c0 + 2) * NHW];
            v2f bA; bA[0] = a0 * a0; bA[1] = a1 * a1;
            accA = __builtin_amdgcn_wmma_f32_16x16x4_f32(
                false, ones, false, bA, (short)0, accA, false, false);

            // Chunk 2: channels c0+4..c0+7.
            float b0 = p[(size_t)(c0 + 4) * NHW];
            float b1 = p[(size_t)(c0 + 6) * NHW];
            v2f bB; bB[0] = b0 * b0; bB[1] = b1 * b1;
            accB = __builtin_amdgcn_wmma_f32_16x16x4_f32(
                false, ones, false, bB, (short)0, accB, false, false);
        }
        // Tail loop: last PFD channels, no prefetch.
        #pragma unroll 4
        for (int c0 = NCM; c0 < NC; c0 += 8) {
            float a0 = p[(size_t)(c0    ) * NHW];
            float a1 = p[(size_t)(c0 + 2) * NHW];
            v2f bA; bA[0] = a0 * a0; bA[1] = a1 * a1;
            accA = __builtin_amdgcn_wmma_f32_16x16x4_f32(
                false, ones, false, bA, (short)0, accA, false, false);

            float b0 = p[(size_t)(c0 + 4) * NHW];
            float b1 = p[(size_t)(c0 + 6) * NHW];
            v2f bB; bB[0] = b0 * b0; bB[1] = b1 * b1;
            accB = __builtin_amdgcn_wmma_f32_16x16x4_f32(
                false, ones, false, bB, (short)0, accB, false, false);
        }

        // Every D row equals the column sum; c[0] on lanes 0-15 is (M=0, N=nsub).
        float colsum = accA[0] + accB[0];
        if (lane < 16) {
            lds_f[col] = colsum;
        }
    }
    __syncthreads();

    // Block reduction over the 288 f values: sum(f) and sum(f^2).
    float s1 = 0.0f, s2 = 0.0f;
    for (int i = tid; i < NHW; i += 128) {
        float v = lds_f[i];
        s1 += v;
        s2 += v * v;
    }
    red1[tid] = s1;
    red2[tid] = s2;
    __syncthreads();
    for (int s = 64; s > 0; s >>= 1) {
        if (tid < s) {
            red1[tid] += red1[tid + s];
            red2[tid] += red2[tid + s];
        }
        __syncthreads();
    }
    if (tid == 0) {
        float norm = sqrtf(red2[0]);
        float inv  = 1.0f / fmaxf(norm, EPSF);
        inv_lds = inv;
        sfn_out[bp] = red1[0] * inv;   // sum of normalized f for this (b,p)
    }
    __syncthreads();

    const float inv = inv_lds;
    float* out = fn_out + (size_t)bp * NHW;
    for (int i = tid; i < NHW; i += 128) {
        out[i] = lds_f[i] * inv;
    }
}

// ---------------------------------------------------------------------------
// Kernel B: per batch b, num = sum_{j<k} sum_hw min(fn_j, fn_k),
// den = max(sum_p sfn, eps); emit num/den. Grid: 32 blocks, 256 threads.
// ---------------------------------------------------------------------------
__global__ __launch_bounds__(256) void sep_pairs_kernel(
    const float* __restrict__ fn,   // [NBP, NHW]
    const float* __restrict__ sfn,  // [NBP]
    float* __restrict__ perb)       // [NB]
{
    __shared__ float lds[NP * NHW];   // 2304 floats
    __shared__ float red[256];

    const int tid = threadIdx.x;
    const int b   = blockIdx.x;

    const float* src = fn + (size_t)b * (NP * NHW);
    for (int i = tid; i < NP * NHW; i += 256) {
        lds[i] = src[i];
    }
    __syncthreads();

    float acc = 0.0f;
    #pragma unroll
    for (int j = 0; j < NP - 1; ++j) {
        #pragma unroll
        for (int k = j + 1; k < NP; ++k) {
            const float* fj = &lds[j * NHW];
            const float* fk = &lds[k * NHW];
            for (int r = tid; r < NHW; r += 256) {
                acc += fminf(fj[r], fk[r]);
            }
        }
    }
    red[tid] = acc;
    __syncthreads();
    for (int s = 128; s > 0; s >>= 1) {
        if (tid < s) red[tid] += red[tid + s];
        __syncthreads();
    }
    if (tid == 0) {
        float den = 0.0f;
        #pragma unroll
        for (int p = 0; p < NP; ++p) den += sfn[b * NP + p];
        perb[b] = red[0] / fmaxf(den, EPSF);
    }
}

// ---------------------------------------------------------------------------
// Kernel C: final scalar = sum over 32 batches (single wave32, shfl reduce).
// ---------------------------------------------------------------------------
__global__ __launch_bounds__(32) void sep_final_kernel(
    const float* __restrict__ perb, float* __restrict__ out)
{
    const int lane = threadIdx.x & 31;
    float v = perb[lane];
    #pragma unroll
    for (int off = 16; off > 0; off >>= 1) {
        v += __shfl_down(v, off, 32);
    }
    if (lane == 0) out[0] = v;
}

extern "C" void kernel_launch(void* const* d_in, const int* in_sizes, int n_in,
                              void* d_out, int out_size, void* d_ws, size_t ws_size,
                              hipStream_t stream) {
    (void)in_sizes; (void)n_in; (void)out_size; (void)ws_size;
    const float* x = (const float*)d_in[0];     // [32,8,512,24,12] fp32
    // d_in[1] (targets, int64) is unused by the reference forward pass.
    float* out = (float*)d_out;                  // scalar fp32

    // Workspace layout (floats): fn[256*288] | sfn[256] | perb[32]  (~296 KB)
    float* ws_fn   = (float*)d_ws;
    float* ws_sfn  = ws_fn + (size_t)NBP * NHW;
    float* ws_perb = ws_sfn + NBP;

    sep_sumsq_norm_kernel<<<NBP, 128, 0, stream>>>(x, ws_fn, ws_sfn);
    sep_pairs_kernel<<<NB, 256, 0, stream>>>(ws_fn, ws_sfn, ws_perb);
    sep_final_kernel<<<1, 32, 0, stream>>>(ws_perb, out);
}